// Maximum_Path_Generator_32263794327646
// MI455X (gfx1250) — compile-verified
//
#include <hip/hip_runtime.h>

// Monotonic Alignment Search (Viterbi DP + backtrace) for MI455X / gfx1250.
// One wave32 per batch row; 16 t-values per lane in VGPRs; circular lane
// shuffle for the t-1 shift; async-DMA (GLOBAL_LOAD_ASYNC_TO_LDS) ring
// prefetch of input rows; backtrace decision bits packed into LDS.

#define NEGC (-1e9f)

constexpr int B_ = 32;
constexpr int F_ = 2048;
constexpr int T_ = 512;
constexpr int LANES_ = 32;                 // wave32
constexpr int TPL_ = T_ / LANES_;          // 16 t-values per lane
constexpr int CMP_BYTES_ = F_ * (T_ / 8);  // 128 KiB decision-bit table in LDS

#if defined(__has_builtin)
#  if __has_builtin(__builtin_amdgcn_global_load_async_to_lds_b128) && \
      __has_builtin(__builtin_amdgcn_s_wait_asynccnt)
#    define MAS_ASYNC 1
#  endif
#endif
#ifndef MAS_ASYNC
#  define MAS_ASYNC 0
#endif

#define RING_DEPTH 8                        // rows in flight (8 async instrs per row)
#define RING_FLOATS (2 * T_)                // neg_cent row + mask row per slot
#define RING_BYTES (RING_DEPTH * RING_FLOATS * 4)

#if MAS_ASYNC
#  define SHMEM_BYTES (CMP_BYTES_ + RING_BYTES)
#else
#  define SHMEM_BYTES (CMP_BYTES_)
#endif

// The async-to-LDS builtin takes pointers to 16-byte int vectors:
//   (int __vector(4) __device__*, int __vector(4) __shared__*, imm, imm)
typedef int mas_v4i __attribute__((vector_size(16)));
typedef __attribute__((address_space(1))) mas_v4i mas_v4i_g;  // global
typedef __attribute__((address_space(3))) mas_v4i mas_v4i_l;  // LDS

__device__ __forceinline__ int imax_(int a, int b) { return a > b ? a : b; }
__device__ __forceinline__ int imin_(int a, int b) { return a < b ? a : b; }

__launch_bounds__(LANES_)
__global__ void mas_dp_kernel(const float* __restrict__ nc,
                              const float* __restrict__ mk,
                              float* __restrict__ out) {
  const int b = blockIdx.x;
  const int lane = threadIdx.x;
  extern __shared__ char smem[];
  unsigned short* cmpTab = reinterpret_cast<unsigned short*>(smem);
#if MAS_ASYNC
  float* ring = reinterpret_cast<float*>(smem + CMP_BYTES_);
#endif
  const size_t base = (size_t)b * F_ * T_;
  const float* ncB = nc + base;
  const float* mkB = mk + base;
  const int tBase = lane * TPL_;

  // ---- token/feature lengths from the mask (tl = sum_t mask[b,0,t], fl = sum_f mask[b,f,0]) ----
  float s0 = 0.f, s1 = 0.f;
#pragma unroll
  for (int j = 0; j < TPL_; ++j) s0 += mkB[tBase + j];
  for (int f = lane; f < F_; f += LANES_) s1 += mkB[(size_t)f * T_];
#pragma unroll
  for (int o = 16; o; o >>= 1) {
    s0 += __shfl_xor(s0, o, LANES_);
    s1 += __shfl_xor(s1, o, LANES_);
  }
  const int tl = (int)(s0 + 0.5f);
  const int fl = (int)(s1 + 0.5f);

  // ---- forward DP state: one Q row (T=512 floats) striped 16-per-lane ----
  float Q[TPL_];
#pragma unroll
  for (int j = 0; j < TPL_; ++j) Q[j] = 0.f;

  // One DP step: consumes row f of (nc,mk) from registers, updates Q in place,
  // and writes the backtrace comparison bits (Q[f-1,t] < Q[f-1,(t-1)%T]) to LDS.
  auto step = [&](int f, const float* xn, const float* xm) {
    const float carry = __shfl(Q[TPL_ - 1], (lane + LANES_ - 1) & (LANES_ - 1), LANES_);
    const int lower = imax_(0, tl + f - fl);
    const int upper = imin_(tl, f + 1);
    unsigned bits = 0u;
    float Qn[TPL_];
#pragma unroll
    for (int j = 0; j < TPL_; ++j) {
      const int t = tBase + j;
      const float praw = (j == 0) ? carry : Q[j - 1];   // Q[f-1, (t-1) mod T], circular
      bits |= (Q[j] < praw) ? (1u << j) : 0u;           // backtrace decision bit for row f
      const float cur = (t == f) ? NEGC : Q[j];
      float prv = praw;
      if (j == 0) prv = (t == 0) ? ((f == 0) ? 0.f : NEGC) : praw;
      const float xv = xn[j] * xm[j];
      const bool band = (t >= lower) && (t < upper);
      Qn[j] = band ? (xv + fmaxf(cur, prv)) : xv;
    }
#pragma unroll
    for (int j = 0; j < TPL_; ++j) Q[j] = Qn[j];
    if (f > 0) cmpTab[f * LANES_ + lane] = (unsigned short)bits;  // row 0 wraps: filled after loop
  };

#if MAS_ASYNC
  // ---- async-DMA front end: ring of RING_DEPTH rows in LDS, 8 async b128 per row ----
  auto issueRow = [&](int f) {
    const int s = f & (RING_DEPTH - 1);
    const float* gn = ncB + (size_t)f * T_ + tBase;
    const float* gm = mkB + (size_t)f * T_ + tBase;
    float* ln = ring + s * RING_FLOATS + tBase;
    float* lm = ring + s * RING_FLOATS + T_ + tBase;
#pragma unroll
    for (int k = 0; k < 4; ++k) {
      __builtin_amdgcn_global_load_async_to_lds_b128(
          (mas_v4i_g*)(gn + 4 * k), (mas_v4i_l*)(ln + 4 * k), 0, 0);
      __builtin_amdgcn_global_load_async_to_lds_b128(
          (mas_v4i_g*)(gm + 4 * k), (mas_v4i_l*)(lm + 4 * k), 0, 0);
    }
  };
  for (int f = 0; f < RING_DEPTH; ++f) issueRow(f);
  for (int f = 0; f < F_; ++f) {
    __builtin_amdgcn_s_wait_asynccnt(8 * (RING_DEPTH - 1));  // oldest row landed in LDS
    const int s = f & (RING_DEPTH - 1);
    const float4* pn = reinterpret_cast<const float4*>(ring + s * RING_FLOATS + tBase);
    const float4* pm = reinterpret_cast<const float4*>(ring + s * RING_FLOATS + T_ + tBase);
    float xn[TPL_], xm[TPL_];
#pragma unroll
    for (int k = 0; k < 4; ++k) {
      const float4 v = pn[k], w = pm[k];
      xn[4 * k + 0] = v.x; xn[4 * k + 1] = v.y; xn[4 * k + 2] = v.z; xn[4 * k + 3] = v.w;
      xm[4 * k + 0] = w.x; xm[4 * k + 1] = w.y; xm[4 * k + 2] = w.z; xm[4 * k + 3] = w.w;
    }
    step(f, xn, xm);
    if (f + RING_DEPTH < F_) issueRow(f + RING_DEPTH);
  }
#else
  // ---- fallback: register double-buffer + L2 prefetch hints ----
  auto loadRow = [&](int f, float* xn, float* xm) {
    const float4* pn = reinterpret_cast<const float4*>(ncB + (size_t)f * T_ + tBase);
    const float4* pm = reinterpret_cast<const float4*>(mkB + (size_t)f * T_ + tBase);
#pragma unroll
    for (int k = 0; k < 4; ++k) {
      const float4 v = pn[k], w = pm[k];
      xn[4 * k + 0] = v.x; xn[4 * k + 1] = v.y; xn[4 * k + 2] = v.z; xn[4 * k + 3] = v.w;
      xm[4 * k + 0] = w.x; xm[4 * k + 1] = w.y; xm[4 * k + 2] = w.z; xm[4 * k + 3] = w.w;
    }
  };
  float xa[2][TPL_], xb[2][TPL_];
  loadRow(0, xa[0], xb[0]);
  loadRow(1, xa[1], xb[1]);
  for (int f = 0; f < F_; ++f) {
    const int pf = f + 24;
    if (pf < F_) {
      __builtin_prefetch(ncB + (size_t)pf * T_ + tBase, 0, 3);  // global_prefetch_b8
      __builtin_prefetch(mkB + (size_t)pf * T_ + tBase, 0, 3);
    }
    step(f, xa[f & 1], xb[f & 1]);
    if (f + 2 < F_) loadRow(f + 2, xa[f & 1], xb[f & 1]);
  }
#endif

  // ---- decision bits for f = 0 use Q[(0-1) mod F] = Q[F-1] (numpy wraparound) ----
  {
    const float carry = __shfl(Q[TPL_ - 1], (lane + LANES_ - 1) & (LANES_ - 1), LANES_);
    unsigned bits = 0u;
#pragma unroll
    for (int j = 0; j < TPL_; ++j) {
      const float praw = (j == 0) ? carry : Q[j - 1];
      bits |= (Q[j] < praw) ? (1u << j) : 0u;
    }
    cmpTab[lane] = (unsigned short)bits;
  }

  // ---- serial backtrace: 2048 dependent LDS bit lookups (per-batch, lane 0) ----
  if (lane == 0) {
    int t = tl - 1;
    float* ob = out + base;
    for (int f = F_ - 1; f >= 0; --f) {
      const int tm = t & (T_ - 1);  // numpy mod-T wrap (T is a power of two)
      const unsigned w = cmpTab[f * LANES_ + (tm >> 4)];
      const int dec = (((t != 0) && (t == f)) || ((w >> (tm & 15)) & 1u)) ? 1 : 0;
      if (f < fl) {
        ob[(size_t)f * T_ + tm] = 1.0f;
        t -= dec;
      }
    }
  }
}

extern "C" void kernel_launch(void* const* d_in, const int* in_sizes, int n_in,
                              void* d_out, int out_size, void* d_ws, size_t ws_size,
                              hipStream_t stream) {
  (void)in_sizes; (void)n_in; (void)d_ws; (void)ws_size;
  const float* nc = (const float*)d_in[0];
  const float* mk = (const float*)d_in[1];
  float* out = (float*)d_out;

  // Zero the one-hot output (harness poisons it; 128 MB at 23.3 TB/s ~ 5.5 us).
  (void)hipMemsetAsync(out, 0, (size_t)out_size * sizeof(float), stream);

  // 128 KiB decision table (+ async ring) of dynamic LDS per workgroup; WGP has 320 KiB.
  (void)hipFuncSetAttribute((const void*)mas_dp_kernel,
                            hipFuncAttributeMaxDynamicSharedMemorySize, (int)SHMEM_BYTES);

  mas_dp_kernel<<<B_, LANES_, SHMEM_BYTES, stream>>>(nc, mk, out);
}